// CritiGraph_38680475468248
// MI455X (gfx1250) — compile-verified
//
#include <hip/hip_runtime.h>
#include <hip/hip_bf16.h>
#include <stdint.h>

// ---------------------------------------------------------------------------
// CritiGraph distance kernel for MI455X (gfx1250, wave32).
//
// Roofline: 67.2 MB of f32 output @ 23.3 TB/s => ~2.9us store floor.
// Dominant compute: 8.4M threefry2x32 evals (~70 VALU ops each) => VALU-bound
// ~2-3x the store floor.  Optimizations:
//   * each random mask evaluated exactly once; the "-result" mirror candidate
//     is derived algebraically (dist(-x,ori) = x==0 ? d : -d),
//   * threefry skipped for tree level i==0 (mask is identically 0),
//   * all arithmetic in int32; only the low dword of each int64 table entry
//     is gathered (values fit in [-65536,65535]),
//   * jax key schedule folded into constexpr (compile-time threefry),
//   * distance folded to a single FMA + sign-bit XORs,
//   * output streamed with non-temporal stores (write-once, 67 MB),
//   * CDNA5 async global->LDS DMA (global_load_async_to_lds_b32/b128 +
//     s_wait_asynccnt) for the per-row gather and the shared inverse
//     permutation table.
// No WMMA: there is no K-reduction / matmul structure anywhere in this op;
// matrix pipes cannot express XOR/clz pointwise work.
// ---------------------------------------------------------------------------

#define H_    16
#define TPD   16
#define KD    16
#define NCAND 513          // 2*H*K + 1

struct U2 { unsigned a, b; };

__host__ __device__ constexpr unsigned rotl32(unsigned x, int r) {
  return (x << r) | (x >> (32 - r));
}

// Threefry-2x32, 20 rounds, exactly as in jax._src.prng.
__host__ __device__ constexpr U2 threefry2x32(unsigned k0, unsigned k1,
                                              unsigned c0, unsigned c1) {
  unsigned ks2 = k0 ^ k1 ^ 0x1BD11BDAu;
  unsigned x0 = c0 + k0, x1 = c1 + k1;
  x0 += x1; x1 = rotl32(x1, 13); x1 ^= x0;
  x0 += x1; x1 = rotl32(x1, 15); x1 ^= x0;
  x0 += x1; x1 = rotl32(x1, 26); x1 ^= x0;
  x0 += x1; x1 = rotl32(x1,  6); x1 ^= x0;
  x0 += k1;  x1 += ks2 + 1u;
  x0 += x1; x1 = rotl32(x1, 17); x1 ^= x0;
  x0 += x1; x1 = rotl32(x1, 29); x1 ^= x0;
  x0 += x1; x1 = rotl32(x1, 16); x1 ^= x0;
  x0 += x1; x1 = rotl32(x1, 24); x1 ^= x0;
  x0 += ks2; x1 += k0 + 2u;
  x0 += x1; x1 = rotl32(x1, 13); x1 ^= x0;
  x0 += x1; x1 = rotl32(x1, 15); x1 ^= x0;
  x0 += x1; x1 = rotl32(x1, 26); x1 ^= x0;
  x0 += x1; x1 = rotl32(x1,  6); x1 ^= x0;
  x0 += k0;  x1 += k1 + 3u;
  x0 += x1; x1 = rotl32(x1, 17); x1 ^= x0;
  x0 += x1; x1 = rotl32(x1, 29); x1 ^= x0;
  x0 += x1; x1 = rotl32(x1, 16); x1 ^= x0;
  x0 += x1; x1 = rotl32(x1, 24); x1 ^= x0;
  x0 += k1;  x1 += ks2 + 4u;
  x0 += x1; x1 = rotl32(x1, 13); x1 ^= x0;
  x0 += x1; x1 = rotl32(x1, 15); x1 ^= x0;
  x0 += x1; x1 = rotl32(x1, 26); x1 ^= x0;
  x0 += x1; x1 = rotl32(x1,  6); x1 ^= x0;
  x0 += ks2; x1 += k0 + 5u;
  return {x0, x1};
}

// Compile-time jax key schedule: key(42) = (0,42); split -> (kmask, kperm);
// split(kperm) -> (_, shuffle subkey).  All folded by constexpr.
constexpr U2 SPL0 = threefry2x32(0u, 42u, 0u, 2u);   // lane 0 of split(key42)
constexpr U2 SPL1 = threefry2x32(0u, 42u, 1u, 3u);   // lane 1
constexpr unsigned KMASK0 = SPL0.a, KMASK1 = SPL1.a; // keys[0]
constexpr unsigned KPERM0 = SPL0.b, KPERM1 = SPL1.b; // keys[1]
constexpr U2 PS0 = threefry2x32(KPERM0, KPERM1, 0u, 2u);
constexpr U2 PS1 = threefry2x32(KPERM0, KPERM1, 1u, 3u);
constexpr unsigned SUBK0 = PS0.b, SUBK1 = PS1.b;     // shuffle subkey

// ---------------------------------------------------------------------------
// Kernel 1: reproduce jax.random.permutation(kperm, 513) (1 sort round since
// ceil(3*ln513/ln(2^32-1)) == 1) and write the INVERSE permutation
// (513 ints + 3 pad) into workspace.  Stable sort emulated via unique
// composite keys (bits<<32 | index) and a rank-count sort in LDS.
// ---------------------------------------------------------------------------
__global__ __launch_bounds__(544) void critigraph_perm_kernel(int* __restrict__ inv) {
  __shared__ __align__(16) unsigned long long skey[NCAND];
  __shared__ __align__(16) int sval[NCAND];
  const int i = threadIdx.x;

  if (i < NCAND) {
    // jax random_bits(subkey, 32, (513,)): odd size -> counts padded to 514;
    // lanes m: (m, 257+m) for m<256, lane 256: (256, 0).
    // out[i] = y0(lane i) for i<257 else y1(lane i-257).
    unsigned lane = (i < 257) ? (unsigned)i : (unsigned)(i - 257);
    unsigned c1   = (lane < 256u) ? (257u + lane) : 0u;
    U2 o = threefry2x32(SUBK0, SUBK1, lane, c1);
    unsigned bits = (i < 257) ? o.a : o.b;
    skey[i] = ((unsigned long long)bits << 32) | (unsigned)i;
  }
  __syncthreads();
  if (i < NCAND) {
    const unsigned long long mykey = skey[i];
    int rank = 0;
    for (int j = 0; j < NCAND; ++j) rank += (skey[j] < mykey) ? 1 : 0;
    sval[rank] = i;               // perm[rank] = i  (x was iota)
  }
  __syncthreads();
  if (i < NCAND) {
    inv[sval[i]] = i;             // inverse permutation: candidate -> out slot
  }
  if (i >= NCAND && i < 516) inv[i] = 0;   // pad so b128 async reads are safe
}

// ---------------------------------------------------------------------------
// Kernel 2: main fused kernel.  One block (8 waves) per row t; each half-wave
// owns one candidate's 16 coordinate dims.
// ---------------------------------------------------------------------------
__global__ __launch_bounds__(256) void critigraph_main_kernel(
    const long long* __restrict__ locations,   // (EMB, 16) int64
    const long long* __restrict__ pos_idx,     // (T,)      int64
    const float*     __restrict__ norm,        // (T,)      f32
    float*           __restrict__ out,         // (T, 513, 16) f32
    const int*       __restrict__ inv_ws,      // (516,) inverse permutation
    int T) {
  __shared__ __align__(16) int s_inv[516];   // inverse permutation
  __shared__ __align__(16) int s_ori[32];    // 16 int64 entries as 32 dwords

  const int t   = blockIdx.x;
  const int tid = threadIdx.x;

  // ---- CDNA5 async global->LDS DMA for the gathered row + perm table ----
  const long long pos = pos_idx[t];                    // scalar (s_load)
  const unsigned row_byte = (unsigned)(pos * (TPD * 8));
  if (tid < 32) {                                      // wave 0: 32 dwords
    unsigned lds_off = (unsigned)(size_t)(&s_ori[tid]);
    unsigned voff    = row_byte + (unsigned)tid * 4u;
    asm volatile("global_load_async_to_lds_b32 %0, %1, %2"
                 :: "v"(lds_off), "v"(voff),
                    "s"((unsigned long long)(size_t)locations)
                 : "memory");
  }
  if (tid < 129) {                                     // 129 x b128 = 516 ints
    unsigned lds_off = (unsigned)(size_t)(&s_inv[tid * 4]);
    unsigned voff    = (unsigned)tid * 16u;
    asm volatile("global_load_async_to_lds_b128 %0, %1, %2"
                 :: "v"(lds_off), "v"(voff),
                    "s"((unsigned long long)(size_t)inv_ws)
                 : "memory");
  }
  asm volatile("s_wait_asynccnt 0x0" ::: "memory");
  __syncthreads();

  const int   tp   = tid & 15;
  const int   csub = tid >> 4;            // half-wave -> one candidate
  const float nrm  = norm[t];             // scalar load
  const float nrm16 = nrm * 0.0625f;      // norm / H
  const int   ori  = s_ori[2 * tp];       // low dword of int64 (value fits)
  const int   aori = (ori < 0) ? -ori : ori;
  const size_t base = (size_t)t * (NCAND * TPD);

  for (int c = csub; c < 257; c += 16) {
    int x;
    if (c == 256) {
      x = ori;                            // the "ori" candidate
    } else {
      const int lvl = c >> 4;             // tree level i in [0,16)
      const int kk  = c & 15;             // k in [0,16)
      int mask = 0;
      if (lvl > 0) {
        // jax randint(.., 0, 65536, int64) with span 2^16 == low 16 bits of
        // the low word of the 64-bit draw; low word = y1 of lane idx with
        // counters (idx, idx + 2^23).  Flat index over (H, T, K, TP).
        const unsigned idx =
            (((unsigned)lvl * (unsigned)T + (unsigned)t) * KD + (unsigned)kk) * TPD +
            (unsigned)tp;
        const U2 o = threefry2x32(KMASK0, KMASK1, idx, idx + (1u << 23));
        mask = (int)(o.b & ((1u << lvl) - 1u));
      }
      x = (ori ^ (1 << lvl)) ^ mask;      // flipped ^ rmask
    }

    // distance(x, ori) = sg * (1 - bitlen(|x|^|ori| + 1)/16) * norm
    //                  = sg * fma(-e, norm/16, norm)
    const int ax = (x < 0) ? -x : x;
    const unsigned v = (unsigned)(ax ^ aori) + 1u;   // in [1, 2^17]: exact
    const int e = 32 - __clz((int)v);                // frexp exponent
    const float dmag = __builtin_fmaf(-(float)e, nrm16, nrm);

    // sign via sign-bit XOR: sg = -1 iff sign bits of x and ori differ
    unsigned dbits = __float_as_uint(dmag) ^ ((unsigned)(x ^ ori) & 0x80000000u);
    const int j1 = s_inv[c];
    __builtin_nontemporal_store(__uint_as_float(dbits),
                                &out[base + (size_t)j1 * TPD + tp]);

    if (c < 256) {
      // mirror candidate -result[c] at pre-perm slot c+257:
      // |x| unchanged, sign flips unless x == 0.
      unsigned flip = (x == 0) ? 0u : 0x80000000u;
      const int j2 = s_inv[c + 257];
      __builtin_nontemporal_store(__uint_as_float(dbits ^ flip),
                                  &out[base + (size_t)j2 * TPD + tp]);
    }
  }
}

// ---------------------------------------------------------------------------
extern "C" void kernel_launch(void* const* d_in, const int* in_sizes, int n_in,
                              void* d_out, int out_size, void* d_ws, size_t ws_size,
                              hipStream_t stream) {
  (void)n_in; (void)out_size; (void)ws_size;
  const long long* locations = (const long long*)d_in[0];  // int64 (EMB,16)
  const long long* pos_idx   = (const long long*)d_in[1];  // int64 (T,)
  const float*     norm      = (const float*)d_in[2];      // f32   (T,1,1)
  float*           out       = (float*)d_out;
  int*             inv       = (int*)d_ws;                 // 516 ints scratch

  const int T = in_sizes[2];                               // 2048

  critigraph_perm_kernel<<<1, 544, 0, stream>>>(inv);
  critigraph_main_kernel<<<T, 256, 0, stream>>>(locations, pos_idx, norm, out,
                                                inv, T);
}